// PointTransformer_29678224016145
// MI455X (gfx1250) — compile-verified
//
#include <hip/hip_runtime.h>
#include <stdint.h>

#define K16 16

typedef float v2f __attribute__((ext_vector_type(2)));
typedef float v8f __attribute__((ext_vector_type(8)));
typedef unsigned int u32x4 __attribute__((ext_vector_type(4)));
typedef int i32x4 __attribute__((ext_vector_type(4)));
typedef int i32x8 __attribute__((ext_vector_type(8)));

// ---------------------------------------------------------------------------
// fp32 WMMA GEMM: C[M,N] = act(A[M,Kd] @ W[Kd,N] + bias[N] (+ resid))
//
// Block = 64 threads (2 waves). Each wave owns one 16-row M tile; the block
// shares a [Kd x 16*TN] weight tile staged into LDS ONCE via the Tensor Data
// Mover (tensor_load_to_lds, TENSORcnt). Each wave computes TN 16x16 output
// tiles: one A fragment feeds TN independent V_WMMA_F32_16X16X4_F32 per
// K-step (ILP to hide load latency). Kd%4 handled by a scalar tail (K=3 and
// K=D+3 layers). Requires M%16==0 && N%(16*TN)==0 (true for all layers).
//
// Fragment layouts per CDNA5 ISA 7.12.2 (wave32):
//   A 16x4 :  lane L: row m0+(L&15), K pair 2*(L>>4)..+1   -> v2f
//   B 4x16 :  lane L: col n +(L&15), K rows 2*(L>>4), +1   -> v2f (from LDS)
//   C 16x16:  acc[r] lane L -> row m0 + r + 8*(L>>4), col n+(L&15)
// ---------------------------------------------------------------------------
template <int TN>
__global__ void k_gemm_t(const float* __restrict__ A, const float* __restrict__ W,
                         const float* __restrict__ bias, const float* __restrict__ resid,
                         float* __restrict__ C, int M, int Kd, int N, int act)
{
  constexpr int W16 = 16 * TN;
  extern __shared__ float sW[];  // [Kd][W16]

  const int n0   = blockIdx.y * W16;
  const int wave = threadIdx.x >> 5;
  const int lane = threadIdx.x & 31;
  const int half = lane >> 4;
  const int l16  = lane & 15;
  const int m0   = (blockIdx.x * 2 + wave) * 16;

  // ---- stage weight tile [Kd x W16] (rows k, cols n0..n0+W16) into LDS ----
#if __has_builtin(__builtin_amdgcn_tensor_load_to_lds)
  if (wave == 0) {
    // Tensor DMA descriptor (ISA 8.3-8.6): 2-D tensor, 4-byte elements.
    const uint64_t ga  = (uint64_t)(uintptr_t)(W + n0);  // tile start
    const uint32_t lds = (uint32_t)(uintptr_t)(&sW[0]);
    const uint32_t td0 = (uint32_t)N;    // tensor_dim0 (row length)
    const uint32_t td1 = (uint32_t)Kd;   // tensor_dim1 (rows)
    const uint32_t st0 = (uint32_t)N;    // tensor_dim0_stride
    u32x4 g0;
    g0[0] = 1u;                                   // count=1 (valid descriptor)
    g0[1] = lds;                                  // lds_addr
    g0[2] = (uint32_t)ga;                         // global_addr[31:0]
    g0[3] = (uint32_t)((ga >> 32) & 0x1FFFFFFu) | (2u << 30);  // addr hi | type=2
    i32x8 g1;
    g1[0] = (int)(2u << 16);                      // data_size=2 (4B), mask=0
    g1[1] = (int)((td0 & 0xFFFFu) << 16);         // tensor_dim0[15:0]
    g1[2] = (int)(((td0 >> 16) & 0xFFFFu) | ((td1 & 0xFFFFu) << 16));
    g1[3] = (int)(((td1 >> 16) & 0xFFFFu) | ((uint32_t)W16 << 16));  // tile_dim0
    g1[4] = (int)(td1 & 0xFFFFu);                 // tile_dim1=Kd, tile_dim2=0
    g1[5] = (int)st0;                             // tensor_dim0_stride[31:0]
    g1[6] = 0;                                    // stride0[47:32]=0, stride1 lo=0
    g1[7] = 0;
    i32x4 g2 = {0, 0, 0, 0};
    i32x4 g3 = {0, 0, 0, 0};
    i32x8 g4 = {0, 0, 0, 0, 0, 0, 0, 0};          // unused (6-arg builtin form)
    __builtin_amdgcn_tensor_load_to_lds(g0, g1, g2, g3, g4, 0);
#if __has_builtin(__builtin_amdgcn_s_wait_tensorcnt)
    __builtin_amdgcn_s_wait_tensorcnt((short)0);
#else
    asm volatile("s_wait_tensorcnt 0x0" ::: "memory");
#endif
  }
#else
  for (int t = threadIdx.x; t < Kd * W16; t += blockDim.x)
    sW[t] = W[(size_t)(t / W16) * N + n0 + (t % W16)];
#endif
  __syncthreads();

  if (m0 < M) {
    v8f acc[TN];
#pragma unroll
    for (int t = 0; t < TN; ++t) acc[t] = (v8f){0.f,0.f,0.f,0.f,0.f,0.f,0.f,0.f};

    const float* ap = A + (size_t)(m0 + l16) * Kd + 2 * half;
    const float* sp = sW + (2 * half) * W16 + l16;
    int kk = 0;
    for (; kk + 4 <= Kd; kk += 4) {
      v2f a;
      a.x = ap[0];
      a.y = ap[1];
#pragma unroll
      for (int t = 0; t < TN; ++t) {
        v2f b;
        b.x = sp[t * 16];
        b.y = sp[W16 + t * 16];
        acc[t] = __builtin_amdgcn_wmma_f32_16x16x4_f32(false, a, false, b,
                                                       (short)0, acc[t],
                                                       false, false);
      }
      ap += 4;
      sp += 4 * W16;
    }
    for (; kk < Kd; ++kk) {  // scalar K tail (Kd%4)
#pragma unroll
      for (int t = 0; t < TN; ++t) {
        float bval = sW[(size_t)kk * W16 + t * 16 + l16];
#pragma unroll
        for (int r = 0; r < 8; ++r)
          acc[t][r] += A[(size_t)(m0 + r + 8 * half) * Kd + kk] * bval;
      }
    }

#pragma unroll
    for (int t = 0; t < TN; ++t) {
      const int nn = n0 + t * 16 + l16;
      float bv = bias ? bias[nn] : 0.0f;
#pragma unroll
      for (int r = 0; r < 8; ++r) {
        int m = m0 + r + 8 * half;
        float v = acc[t][r] + bv;
        if (resid) v += resid[(size_t)m * N + nn];
        if (act) v = fmaxf(v, 0.0f);
        C[(size_t)m * N + nn] = v;
      }
    }
  }
}

// ---------------------------------------------------------------------------
// KNN: for each of M query points, top-16 nearest (squared dist) among Ncand
// candidates. Candidates staged through LDS in 256-point chunks via async
// global->LDS copies (ASYNCcnt path); each thread keeps a sorted register-
// resident top-16 (unrolled insertion network).
// ---------------------------------------------------------------------------
#define KNN_BLK 128
#define KNN_CH  256

__device__ __forceinline__ void async_ld_b32(uint32_t lds_off, const float* g)
{
  asm volatile("global_load_async_to_lds_b32 %0, %1, off"
               :: "v"(lds_off), "v"(g) : "memory");
}
__device__ __forceinline__ void wait_asynccnt0()
{
  asm volatile("s_wait_asynccnt 0" ::: "memory");
}

__global__ void k_knn(const float* __restrict__ qpos, const float* __restrict__ cand,
                      int* __restrict__ out, int M, int Ncand)
{
  __shared__ float sc[KNN_CH * 3];
  const int  m      = blockIdx.x * KNN_BLK + threadIdx.x;
  const bool active = (m < M);
  const int  mc     = active ? m : (M - 1);
  const float qx = qpos[mc * 3 + 0];
  const float qy = qpos[mc * 3 + 1];
  const float qz = qpos[mc * 3 + 2];

  float bd[K16];
  int   bi[K16];
#pragma unroll
  for (int i = 0; i < K16; ++i) { bd[i] = 3.0e38f; bi[i] = 0; }

  for (int c0 = 0; c0 < Ncand; c0 += KNN_CH) {
    const int rem = Ncand - c0;
    const int lim = rem < KNN_CH ? rem : KNN_CH;
    const float* src = cand + (size_t)c0 * 3;
    for (int t = threadIdx.x; t < lim * 3; t += KNN_BLK)
      async_ld_b32((uint32_t)(uintptr_t)(&sc[t]), src + t);
    wait_asynccnt0();
    __syncthreads();

    for (int j = 0; j < lim; ++j) {
      float dx = sc[j * 3 + 0] - qx;
      float dy = sc[j * 3 + 1] - qy;
      float dz = sc[j * 3 + 2] - qz;
      float d  = dx * dx + dy * dy + dz * dz;
      if (d < bd[K16 - 1]) {
        bd[K16 - 1] = d;
        bi[K16 - 1] = c0 + j;
#pragma unroll
        for (int i2 = K16 - 1; i2 > 0; --i2) {
          if (bd[i2] < bd[i2 - 1]) {
            float tv = bd[i2]; bd[i2] = bd[i2 - 1]; bd[i2 - 1] = tv;
            int   ti = bi[i2]; bi[i2] = bi[i2 - 1]; bi[i2 - 1] = ti;
          }
        }
      }
    }
    __syncthreads();
  }

  if (active) {
#pragma unroll
    for (int i = 0; i < K16; ++i) out[(size_t)m * K16 + i] = bi[i];
  }
}

// ---------------------------------------------------------------------------
// Farthest point sampling (one block of 256 threads per call, per batch).
// Matches jnp.argmax first-occurrence tie-breaking via (d >, idx <) reduce.
// ---------------------------------------------------------------------------
__global__ void k_fps(const float* __restrict__ pos, int* __restrict__ cent,
                      int N, int S)
{
  const int t = threadIdx.x;  // 256 threads
  __shared__ float sv[256];
  __shared__ int   si[256];
  __shared__ float lp[3];

  float dmin[16], px[16], py[16], pz[16];
#pragma unroll
  for (int i = 0; i < 16; ++i) {
    int p = t + i * 256;
    if (p < N) {
      px[i] = pos[p * 3 + 0];
      py[i] = pos[p * 3 + 1];
      pz[i] = pos[p * 3 + 2];
      dmin[i] = 1.0e10f;
    } else {
      px[i] = py[i] = pz[i] = 0.f;
      dmin[i] = -1.0f;
    }
  }
  if (t == 0) cent[0] = 0;
  int last = 0;

  for (int s = 1; s < S; ++s) {
    if (t == 0) {
      lp[0] = pos[last * 3 + 0];
      lp[1] = pos[last * 3 + 1];
      lp[2] = pos[last * 3 + 2];
    }
    __syncthreads();
    float lx = lp[0], ly = lp[1], lz = lp[2];
    float bdv = -2.0f; int biv = 0;
#pragma unroll
    for (int i = 0; i < 16; ++i) {
      int p = t + i * 256;
      if (p < N) {
        float dx = px[i] - lx, dy = py[i] - ly, dz = pz[i] - lz;
        float d = fminf(dmin[i], dx * dx + dy * dy + dz * dz);
        dmin[i] = d;
        if (d > bdv) { bdv = d; biv = p; }
      }
    }
    sv[t] = bdv; si[t] = biv;
    __syncthreads();
    for (int off = 128; off > 0; off >>= 1) {
      if (t < off) {
        if (sv[t + off] > sv[t] ||
            (sv[t + off] == sv[t] && si[t + off] < si[t])) {
          sv[t] = sv[t + off];
          si[t] = si[t + off];
        }
      }
      __syncthreads();
    }
    last = si[0];
    if (t == 0) cent[s] = last;
    __syncthreads();
  }
}

// ------------------------- elementwise helper kernels ----------------------
__global__ void k_gather(const float* __restrict__ src, const int* __restrict__ knn,
                         float* __restrict__ out, int total, int D)
{
  int i = blockIdx.x * blockDim.x + threadIdx.x;
  if (i >= total) return;
  int d = i % D, mk = i / D;
  out[i] = src[(size_t)knn[mk] * D + d];
}

__global__ void k_relpos(const float* __restrict__ qpos, const float* __restrict__ cand,
                         const int* __restrict__ knn, float* __restrict__ out, int total)
{
  int i = blockIdx.x * blockDim.x + threadIdx.x;
  if (i >= total) return;
  int c = i % 3, mk = i / 3, m = mk / K16;
  out[i] = qpos[m * 3 + c] - cand[(size_t)knn[mk] * 3 + c];
}

// s = q_broadcast - s + pe  (in place; s initially holds gathered K-features)
__global__ void k_combine(const float* __restrict__ q, float* __restrict__ s,
                          const float* __restrict__ pe, int total, int D)
{
  int i = blockIdx.x * blockDim.x + threadIdx.x;
  if (i >= total) return;
  int d = i % D, m = i / (D * K16);
  s[i] = q[(size_t)m * D + d] - s[i] + pe[i];
}

// res[m,d] = softmax_k(attn[m,k,d]*scale) . (v[m,k,d] + pe[m,k,d])
__global__ void k_attn_reduce(const float* __restrict__ attn, const float* __restrict__ v,
                              const float* __restrict__ pe, float* __restrict__ res,
                              int total, int D, float scale)
{
  int j = blockIdx.x * blockDim.x + threadIdx.x;
  if (j >= total) return;
  int d = j % D, m = j / D;
  size_t base = ((size_t)m * K16) * D + d;
  float mx = -3.0e38f;
#pragma unroll
  for (int k = 0; k < K16; ++k)
    mx = fmaxf(mx, attn[base + (size_t)k * D] * scale);
  float sum = 0.f, ws = 0.f;
#pragma unroll
  for (int k = 0; k < K16; ++k) {
    size_t o = base + (size_t)k * D;
    float e = expf(attn[o] * scale - mx);
    sum += e;
    ws  += e * (v[o] + pe[o]);
  }
  res[j] = ws / sum;
}

__global__ void k_maxpool(const float* __restrict__ h, float* __restrict__ out,
                          int total, int D)
{
  int j = blockIdx.x * blockDim.x + threadIdx.x;
  if (j >= total) return;
  int d = j % D, m = j / D;
  size_t base = ((size_t)m * K16) * D + d;
  float mx = -3.0e38f;
#pragma unroll
  for (int k = 0; k < K16; ++k)
    mx = fmaxf(mx, h[base + (size_t)k * D]);
  out[j] = mx;
}

// cat[m,k,:] = [ feat[knn[m,k], 0:Dp] | pos[knn[m,k]] - cpos[m] ]
__global__ void k_concat(const float* __restrict__ feat, const float* __restrict__ pos,
                         const float* __restrict__ cpos, const int* __restrict__ knn,
                         float* __restrict__ out, int total, int Dp)
{
  int i = blockIdx.x * blockDim.x + threadIdx.x;
  if (i >= total) return;
  int Din = Dp + 3;
  int c = i % Din, mk = i / Din, m = mk / K16;
  int idx = knn[mk];
  out[i] = (c < Dp) ? feat[(size_t)idx * Dp + c]
                    : pos[(size_t)idx * 3 + (c - Dp)] - cpos[(size_t)m * 3 + (c - Dp)];
}

__global__ void k_centpos(const float* __restrict__ pos, const int* __restrict__ cent,
                          float* __restrict__ cpos, int total)
{
  int i = blockIdx.x * blockDim.x + threadIdx.x;
  if (i >= total) return;
  int c = i % 3, s = i / 3;
  cpos[i] = pos[(size_t)cent[s] * 3 + c];
}

// --------------------------- host-side orchestration -----------------------
struct PTB {
  const float *W1, *b1, *Wq, *Wk, *Wv, *Wd1, *bd1, *Wd2, *bd2,
              *Wg1, *bg1, *Wg2, *bg2, *W2, *b2;
};
struct TD { const float *W1, *b1, *W2, *b2; };

struct Arena {
  char*  base;
  size_t off;
  float* f(size_t n)  { size_t o = (off + 255) & ~(size_t)255; off = o + n * 4; return (float*)(base + o); }
  int*   i32(size_t n){ size_t o = (off + 255) & ~(size_t)255; off = o + n * 4; return (int*)(base + o); }
};

static inline void gemm(hipStream_t st, const float* A, const float* W,
                        const float* bias, const float* resid, float* C,
                        int M, int Kd, int N, int act)
{
  int gx = (M + 31) / 32;
  if (N % 64 == 0) {
    dim3 g(gx, N / 64);
    k_gemm_t<4><<<g, 64, (size_t)Kd * 64 * 4, st>>>(A, W, bias, resid, C, M, Kd, N, act);
  } else {
    dim3 g(gx, N / 32);
    k_gemm_t<2><<<g, 64, (size_t)Kd * 32 * 4, st>>>(A, W, bias, resid, C, M, Kd, N, act);
  }
}
static inline int nb(int total) { return (total + 255) / 256; }

static void run_ptb(hipStream_t st, const PTB& p, const float* h_in,
                    const float* pos, int M, int D, float* h_out, Arena ar)
{
  int*   knn = ar.i32((size_t)M * K16);
  float* rel = ar.f((size_t)M * K16 * 3);
  float* h   = ar.f((size_t)M * D);
  float* q   = ar.f((size_t)M * D);
  float* tf  = ar.f((size_t)M * D);
  float* res = ar.f((size_t)M * D);
  size_t nkd = (size_t)M * K16 * D;
  float* kg  = ar.f(nkd);
  float* vg  = ar.f(nkd);
  float* pe  = ar.f(nkd);
  float* pt  = ar.f(nkd);

  k_knn<<<dim3((M + KNN_BLK - 1) / KNN_BLK), KNN_BLK, 0, st>>>(pos, pos, knn, M, M);
  k_relpos<<<nb(M * K16 * 3), 256, 0, st>>>(pos, pos, knn, rel, M * K16 * 3);

  gemm(st, h_in, p.W1, p.b1, nullptr, h, M, D, D, 0);
  gemm(st, h, p.Wq, nullptr, nullptr, q, M, D, D, 0);
  gemm(st, h, p.Wk, nullptr, nullptr, tf, M, D, D, 0);
  k_gather<<<nb((int)nkd), 256, 0, st>>>(tf, knn, kg, (int)nkd, D);
  gemm(st, h, p.Wv, nullptr, nullptr, tf, M, D, D, 0);
  k_gather<<<nb((int)nkd), 256, 0, st>>>(tf, knn, vg, (int)nkd, D);

  gemm(st, rel, p.Wd1, p.bd1, nullptr, pt, M * K16, 3, D, 1);   // relu
  gemm(st, pt, p.Wd2, p.bd2, nullptr, pe, M * K16, D, D, 0);

  k_combine<<<nb((int)nkd), 256, 0, st>>>(q, kg, pe, (int)nkd, D);  // kg -> s
  gemm(st, kg, p.Wg1, p.bg1, nullptr, pt, M * K16, D, D, 1);        // relu
  gemm(st, pt, p.Wg2, p.bg2, nullptr, kg, M * K16, D, D, 0);        // kg -> attn

  float scale = 1.0f / sqrtf((float)D);
  k_attn_reduce<<<nb(M * D), 256, 0, st>>>(kg, vg, pe, res, M * D, D, scale);

  gemm(st, res, p.W2, p.b2, h_in, h_out, M, D, D, 0);  // + residual x
}

static void run_td(hipStream_t st, const TD& p, const float* pos_in,
                   const float* h_in, int Nin, int Dp, int S,
                   float* pos_out, float* h_out, Arena ar)
{
  const int Dh  = 2 * Dp;
  const int Din = Dp + 3;
  int*   cent = ar.i32(S);
  int*   knn  = ar.i32((size_t)S * K16);
  float* cat  = ar.f((size_t)S * K16 * Din);
  float* t1   = ar.f((size_t)S * K16 * Dh);
  float* t2   = ar.f((size_t)S * K16 * Dh);

  k_fps<<<1, 256, 0, st>>>(pos_in, cent, Nin, S);
  k_centpos<<<nb(S * 3), 256, 0, st>>>(pos_in, cent, pos_out, S * 3);
  k_knn<<<dim3((S + KNN_BLK - 1) / KNN_BLK), KNN_BLK, 0, st>>>(pos_out, pos_in, knn, S, Nin);
  k_concat<<<nb(S * K16 * Din), 256, 0, st>>>(h_in, pos_in, pos_out, knn, cat,
                                              S * K16 * Din, Dp);
  gemm(st, cat, p.W1, p.b1, nullptr, t1, S * K16, Din, Dh, 1);  // relu
  gemm(st, t1, p.W2, p.b2, nullptr, t2, S * K16, Dh, Dh, 1);    // relu
  k_maxpool<<<nb(S * Dh), 256, 0, st>>>(t2, h_out, S * Dh, Dh);
}

static PTB load_ptb(void* const* d_in, int& c)
{
  PTB p;
  p.W1  = (const float*)d_in[c++]; p.b1  = (const float*)d_in[c++];
  p.Wq  = (const float*)d_in[c++]; p.Wk  = (const float*)d_in[c++];
  p.Wv  = (const float*)d_in[c++];
  p.Wd1 = (const float*)d_in[c++]; p.bd1 = (const float*)d_in[c++];
  p.Wd2 = (const float*)d_in[c++]; p.bd2 = (const float*)d_in[c++];
  p.Wg1 = (const float*)d_in[c++]; p.bg1 = (const float*)d_in[c++];
  p.Wg2 = (const float*)d_in[c++]; p.bg2 = (const float*)d_in[c++];
  p.W2  = (const float*)d_in[c++]; p.b2  = (const float*)d_in[c++];
  return p;
}

extern "C" void kernel_launch(void* const* d_in, const int* in_sizes, int n_in,
                              void* d_out, int out_size, void* d_ws, size_t ws_size,
                              hipStream_t stream)
{
  (void)in_sizes; (void)n_in; (void)out_size; (void)ws_size;
  int c = 0;
  const float* x   = (const float*)d_in[c++];
  const float* Wf1 = (const float*)d_in[c++];
  const float* bf1 = (const float*)d_in[c++];
  const float* Wf2 = (const float*)d_in[c++];
  const float* bf2 = (const float*)d_in[c++];
  PTB ptb0 = load_ptb(d_in, c);
  TD td[4];
  for (int i = 0; i < 4; ++i) {
    td[i].W1 = (const float*)d_in[c++];
    td[i].b1 = (const float*)d_in[c++];
    td[i].W2 = (const float*)d_in[c++];
    td[i].b2 = (const float*)d_in[c++];
  }
  PTB tf[4];
  for (int i = 0; i < 4; ++i) tf[i] = load_ptb(d_in, c);

  // persistent per-batch ping-pong buffers carved from the workspace head
  char*  wsb = (char*)d_ws;
  size_t off = 0;
  auto carve = [&](size_t bytes) {
    size_t o = (off + 255) & ~(size_t)255;
    off = o + bytes;
    return wsb + o;
  };
  float* posA = (float*)carve((size_t)4096 * 3 * 4);
  float* posB = (float*)carve((size_t)4096 * 3 * 4);
  float* hA   = (float*)carve((size_t)4096 * 32 * 4);  // max M*D over stages
  float* hB   = (float*)carve((size_t)4096 * 32 * 4);
  Arena ar{wsb, off};

  for (int b = 0; b < 4; ++b) {
    const float* xb = x + (size_t)b * 4096 * 3;
    {  // feature embed: h = relu(x@Wf1+bf1)@Wf2+bf2
      Arena a = ar;
      float* t0 = a.f((size_t)4096 * 32);
      gemm(stream, xb, Wf1, bf1, nullptr, t0, 4096, 3, 32, 1);
      gemm(stream, t0, Wf2, bf2, nullptr, hA, 4096, 32, 32, 0);
    }
    run_ptb(stream, ptb0, hA, xb, 4096, 32, hB, ar);

    const float* pos_cur = xb;
    float* h_cur = hB;
    int n = 4096, D = 32;
    for (int i = 0; i < 4; ++i) {
      int S = n / 4, Dh = D * 2;
      float* pos_new = (i & 1) ? posB : posA;
      float* h_new   = (h_cur == hA) ? hB : hA;
      run_td(stream, td[i], pos_cur, h_cur, n, D, S, pos_new, h_new, ar);
      pos_cur = pos_new;
      h_cur   = h_new;
      float* h_out = (i == 3) ? ((float*)d_out + (size_t)b * 16 * 512)
                              : ((h_cur == hA) ? hB : hA);
      run_ptb(stream, tf[i], h_cur, pos_cur, S, Dh, h_out, ar);
      h_cur = h_out;
      n = S;
      D = Dh;
    }
  }
}